// SubspaceLinopFactory_3977139716861
// MI455X (gfx1250) — compile-verified
//
#include <hip/hip_runtime.h>
#include <hip/hip_bf16.h>

// Problem constants (match reference)
#define A_ 3
#define T_ 32
#define C_ 4
#define R_ 8
#define K_ 1024
#define H_ 64
#define W_ 64
#define N_ 4096  // H_*W_

typedef __attribute__((ext_vector_type(2))) float v2f;
typedef __attribute__((ext_vector_type(8))) float v8f;

// ---------------------------------------------------------------------------
// Kernel 1: w[t,n] = sum_a phi[a,t] * x[a,n]   (fold temporal basis first so
// only trajectory r_t is ever computed per timepoint)
// ---------------------------------------------------------------------------
__global__ __launch_bounds__(256) void k_make_w(const float* __restrict__ x,
                                                const float* __restrict__ phi,
                                                float* __restrict__ w) {
    int i = blockIdx.x * 256 + threadIdx.x;     // T_*N_ = 131072 threads
    int t = i >> 12;                            // /4096
    int n = i & (N_ - 1);
    float acc = 0.f;
#pragma unroll
    for (int a = 0; a < A_; ++a)
        acc = fmaf(phi[a * T_ + t], x[a * N_ + n], acc);
    w[i] = acc;
}

// ---------------------------------------------------------------------------
// Kernel 2: separable trig tables.
//   exc/exs[r,k,x] = cos/sin(trj[r,1,k] * (x-32))   (x-factor of exp(-i phase))
//   eyc/eys[r,k,y] = cos/sin(trj[r,0,k] * (y-32))   (y-factor)
// E = (eyc - i*eys)*(exc - i*exs). Only ~1M sincos instead of 134M.
// ---------------------------------------------------------------------------
__global__ __launch_bounds__(256) void k_tables(const float* __restrict__ trj,
                                                float* __restrict__ exc,
                                                float* __restrict__ exs,
                                                float* __restrict__ eyc,
                                                float* __restrict__ eys) {
    int i = blockIdx.x * 256 + threadIdx.x;     // R_*K_*128 = 1048576 threads
    int slot = i & 127;
    int rk = i >> 7;                            // r*K_ + k
    int r = rk >> 10;
    int k = rk & (K_ - 1);
    float s, c;
    if (slot < W_) {
        float ang = trj[(r * 2 + 1) * K_ + k] * (float)(slot - W_ / 2);
        __sincosf(ang, &s, &c);
        exc[rk * W_ + slot] = c;
        exs[rk * W_ + slot] = s;
    } else {
        int yy = slot - W_;
        float ang = trj[(r * 2 + 0) * K_ + k] * (float)(yy - H_ / 2);
        __sincosf(ang, &s, &c);
        eyc[rk * H_ + yy] = c;
        eys[rk * H_ + yy] = s;
    }
}

// ---------------------------------------------------------------------------
// Kernel 3: WMMA GEMM + y-reduction.
// Per wave: fixed (t, c, 16 k's). For each of 4 y-subtiles accumulate over
// x (Kdim=64, steps of 4) with V_WMMA_F32_16X16X4_F32:
//   accRe[k,y] = sum_x exc[k,x] * (mps[c,y,x]*w[t,y,x])
//   accIm[k,y] = sum_x exs[k,x] * (mps[c,y,x]*w[t,y,x])
// inner = accRe - i*accIm; multiply by (eyc - i*eys)[k,y], reduce over y
// (cross-lane within 16-lane halves), scale by sqrt_dcf, store complex.
// ---------------------------------------------------------------------------
__global__ __launch_bounds__(256) void k_gemm(
    const float* __restrict__ w,    // [T_][N_]
    const float* __restrict__ mps,  // [C_][H_][W_]
    const float* __restrict__ exc, const float* __restrict__ exs,  // [R_][K_][W_]
    const float* __restrict__ eyc, const float* __restrict__ eys,  // [R_][K_][H_]
    const float* __restrict__ dcf,  // [R_][K_]
    const int* __restrict__ sidx,   // [T_]
    float2* __restrict__ out)       // [T_][C_][K_] complex interleaved
{
    const int lane = threadIdx.x & 31;
    const int wave = threadIdx.x >> 5;
    const int unit = blockIdx.x * 8 + wave;   // T_*C_*(K_/16) = 8192 units
    const int t    = unit >> 8;               // 256 units per t
    const int rem  = unit & 255;
    const int c    = rem >> 6;
    const int k0   = (rem & 63) << 4;
    const int r    = sidx[t];

    const int half = lane >> 4;   // which half of the wave
    const int l    = lane & 15;   // lane-in-half: A/C row = k0+l? see below

    const float* exc_r = exc + (size_t)r * K_ * W_;
    const float* exs_r = exs + (size_t)r * K_ * W_;
    const float* eyc_r = eyc + (size_t)r * K_ * H_;
    const float* eys_r = eys + (size_t)r * K_ * H_;
    const float* wt    = w   + (size_t)t * N_;
    const float* mpc   = mps + (size_t)c * N_;

    // hint: stage-2 tables will be needed after the WMMA loops
    __builtin_prefetch(eyc_r + (size_t)(k0 + l) * H_, 0, 1);
    __builtin_prefetch(eys_r + (size_t)(k0 + l) * H_, 0, 1);

    // A (16x4 f32): lanes 0-15 carry K-cols {0,1}, lanes 16-31 carry {2,3},
    // row M = lane&15. So this lane streams exc[k0+l, x0 + 2*half + {0,1}].
    const int arow = k0 + l;

    float outRe[8], outIm[8];
#pragma unroll
    for (int j = 0; j < 8; ++j) { outRe[j] = 0.f; outIm[j] = 0.f; }

    for (int ysub = 0; ysub < 4; ++ysub) {
        const int y = ysub * 16 + l;          // B col / C col for this lane
        v8f accRe = {};
        v8f accIm = {};
#pragma unroll
        for (int kk = 0; kk < 16; ++kk) {
            const int xr = kk * 4 + 2 * half; // even -> 8B aligned v2f loads
            v2f aCos = *(const v2f*)(exc_r + (size_t)arow * W_ + xr);
            v2f aSin = *(const v2f*)(exs_r + (size_t)arow * W_ + xr);
            v2f bm   = *(const v2f*)(mpc   + (size_t)y    * W_ + xr);
            v2f bw   = *(const v2f*)(wt    + (size_t)y    * W_ + xr);
            v2f b = bm * bw;                  // G[x,(c,y)] = mps * w
            // D = A*B + C ; (neg_a, A, neg_b, B, c_mod, C, reuse_a, reuse_b)
            accRe = __builtin_amdgcn_wmma_f32_16x16x4_f32(
                false, aCos, false, b, (short)0, accRe, false, false);
            accIm = __builtin_amdgcn_wmma_f32_16x16x4_f32(
                false, aSin, false, b, (short)0, accIm, false, false);
        }
        // stage 2: multiply tile by (eyc - i*eys)[k,y]; C VGPR j holds rows
        // M=j (lanes 0-15) and M=j+8 (lanes 16-31), col N = lane&15.
#pragma unroll
        for (int j = 0; j < 8; ++j) {
            const int k = k0 + j + 8 * half;
            float ec = eyc_r[(size_t)k * H_ + y];
            float es = eys_r[(size_t)k * H_ + y];
            // (Sre - i*Sim)(ec - i*es) = (Sre*ec - Sim*es) - i(Sre*es + Sim*ec)
            outRe[j] = fmaf(accRe[j], ec, fmaf(-accIm[j], es, outRe[j]));
            outIm[j] = fmaf(accRe[j], es, fmaf( accIm[j], ec, outIm[j]));
        }
    }

    // Reduce over the 16 y-columns (N dim lives in lanes, per 16-lane half;
    // xor masks 1..8 never cross the half boundary on wave32).
#pragma unroll
    for (int j = 0; j < 8; ++j) {
        float re = outRe[j], im = outIm[j];
#pragma unroll
        for (int m = 1; m <= 8; m <<= 1) {
            re += __shfl_xor(re, m, 32);
            im += __shfl_xor(im, m, 32);
        }
        if (l == 0) {
            const int k = k0 + j + 8 * half;
            const float d = dcf[(size_t)r * K_ + k];
            float2 v;
            v.x = re * d;
            v.y = -im * d;   // exp(-i phase): imaginary part is negated sum
            out[((size_t)t * C_ + c) * K_ + k] = v;
        }
    }
}

// ---------------------------------------------------------------------------
extern "C" void kernel_launch(void* const* d_in, const int* in_sizes, int n_in,
                              void* d_out, int out_size, void* d_ws, size_t ws_size,
                              hipStream_t stream) {
    const float* x    = (const float*)d_in[0];   // [A_][H_][W_]
    const float* trj  = (const float*)d_in[1];   // [R_][2][K_]
    const float* phi  = (const float*)d_in[2];   // [A_][T_]
    const float* mps  = (const float*)d_in[3];   // [C_][H_][W_]
    const float* dcf  = (const float*)d_in[4];   // [R_][K_]
    const int*   sidx = (const int*)d_in[5];     // [T_]

    float* ws  = (float*)d_ws;
    float* w   = ws;                                   // T_*N_      = 131072
    float* exc = w   + (size_t)T_ * N_;                // R_*K_*W_   = 524288
    float* exs = exc + (size_t)R_ * K_ * W_;
    float* eyc = exs + (size_t)R_ * K_ * W_;
    float* eys = eyc + (size_t)R_ * K_ * H_;
    // total: 2,228,224 floats ≈ 8.9 MB of workspace

    k_make_w<<<(T_ * N_) / 256, 256, 0, stream>>>(x, phi, w);
    k_tables<<<(R_ * K_ * (W_ + H_)) / 256, 256, 0, stream>>>(trj, exc, exs, eyc, eys);
    k_gemm<<<(T_ * C_ * (K_ / 16)) / 8, 256, 0, stream>>>(
        w, mps, exc, exs, eyc, eys, dcf, sidx, (float2*)d_out);
}